// GRU_83519934038617
// MI455X (gfx1250) — compile-verified
//
#include <hip/hip_runtime.h>
#include <hip/hip_bf16.h>

typedef float v2f __attribute__((ext_vector_type(2)));
typedef float v8f __attribute__((ext_vector_type(8)));
typedef unsigned int v4u __attribute__((ext_vector_type(4)));
typedef int v4i __attribute__((ext_vector_type(4)));
typedef int v8i __attribute__((ext_vector_type(8)));

// Full-precision fp32 WMMA: D = A(16x4) * B(4x16) + C(16x16)
#define WMMA_F32(a, b, c) \
  __builtin_amdgcn_wmma_f32_16x16x4_f32(false, (a), false, (b), (short)0, (c), false, false)

// LDS row stride (floats) after TDM padding: 32 data + 1 pad dword.
#define LDS_STRIDE 33
#define KC 32  // K-chunk per TDM transfer

// ---------------------------------------------------------------------------
// Tensor Data Mover: 2D tile (tile_d1 rows x tile_d0 elems, fp32) from global
// row-major [*, row_stride] into LDS at lds_byte_addr, with 1-dword padding
// every 32 dwords (=> LDS row stride 33 floats).
// D# packing per CDNA5 ISA ch.8 (group0: count/lds/global/type, group1: dims).
// This toolchain exposes the 6-arg builtin: (v4u, v8i, v4i, v4i, v8i, i32).
// ---------------------------------------------------------------------------
static __device__ __forceinline__ void tdm_load_2d(
    unsigned int lds_byte_addr, const float* gptr,
    unsigned int tensor_d0, unsigned int tensor_d1,
    unsigned int tile_d0, unsigned int tile_d1,
    unsigned int row_stride_elems) {
  unsigned long long ga = (unsigned long long)(uintptr_t)gptr;
  v4u g0;
  g0[0] = 1u;                                          // count=1 (user descriptor)
  g0[1] = lds_byte_addr;                               // lds_addr
  g0[2] = (unsigned int)(ga & 0xFFFFFFFFu);            // global_addr[31:0]
  g0[3] = (unsigned int)((ga >> 32) & 0x1FFFFFFu)      // global_addr[56:32]
          | (2u << 30);                                // type=2 ("image")
  unsigned int d0 = (2u << 16)                         // data_size = 4 bytes
                  | (1u << 20)                         // pad_enable
                  | (4u << 22)                         // pad_interval: 32 dwords
                  | (0u << 25);                        // pad_amount: 1 dword
  v8i g1;
  g1[0] = (int)d0;                                     // mask=0, no atomic barrier
  g1[1] = (int)((tensor_d0 & 0xFFFFu) << 16);          // tensor_dim0[15:0] @ bits 63:48
  g1[2] = (int)(((tensor_d0 >> 16) & 0xFFFFu)          // tensor_dim0[31:16]
          | ((tensor_d1 & 0xFFFFu) << 16));            // tensor_dim1[15:0]
  g1[3] = (int)(((tensor_d1 >> 16) & 0xFFFFu)          // tensor_dim1[31:16]
          | ((tile_d0 & 0xFFFFu) << 16));              // tile_dim0
  g1[4] = (int)(tile_d1 & 0xFFFFu);                    // tile_dim1 (tile_dim2=0 -> 2D)
  g1[5] = (int)row_stride_elems;                       // tensor_dim0_stride[31:0]
  g1[6] = 0;                                           // stride[47:32], dim1_stride lo
  g1[7] = 0;
  v4i z4 = {0, 0, 0, 0};
  v8i z8 = {0, 0, 0, 0, 0, 0, 0, 0};
  __builtin_amdgcn_tensor_load_to_lds(g0, g1, z4, z4, z8, 0);
}

// ---------------------------------------------------------------------------
// Fragment loaders.
// A 16x4 fp32 layout: lanes 0-15 -> M=lane, K={k0,k0+1}; lanes 16-31 -> K={k0+2,k0+3}
// B 4x16 fed as W^T: B[k][n] = W[n][k]; per-lane contiguous float2 along K.
// ---------------------------------------------------------------------------
static __device__ __forceinline__ v2f load_a_frag(const float* __restrict__ X, int ld,
                                                  int m0, int k0, int lane) {
  int m  = m0 + (lane & 15);
  int kk = k0 + ((lane >> 4) << 1);
  const float* p = X + (size_t)m * ld + kk;
  v2f r = { p[0], p[1] };
  return r;
}

static __device__ __forceinline__ v2f load_b_frag(const float* __restrict__ W, int ld,
                                                  int n0, int k0, int lane) {
  int n  = n0 + (lane & 15);
  int kk = k0 + ((lane >> 4) << 1);
  const float* p = W + (size_t)n * ld + kk;
  v2f r = { p[0], p[1] };
  return r;
}

// LDS fragment: tile row-major with stride 33 floats, 2 consecutive floats/lane.
static __device__ __forceinline__ v2f lds_frag(const float* S, int r0, int k0, int lane) {
  const float* p = S + (r0 + (lane & 15)) * LDS_STRIDE + k0 + ((lane >> 4) << 1);
  v2f r = { p[0], p[1] };
  return r;
}

static __device__ __forceinline__ float sigmoidf_(float x) {
  return 1.0f / (1.0f + __expf(-x));
}

// ---------------------------------------------------------------------------
// Embedding gather: x[row, :] = table[idx[row], :]
// ---------------------------------------------------------------------------
__global__ void embed_kernel(const int* __restrict__ idx,
                             const float* __restrict__ table,
                             float* __restrict__ x) {
  int row = blockIdx.x;
  int t = idx[row];
  const float* __restrict__ src = table + (size_t)t * 512;
  float* __restrict__ dst = x + (size_t)row * 512;
  for (int i = threadIdx.x; i < 512; i += blockDim.x) dst[i] = src[i];
}

// ---------------------------------------------------------------------------
// TDM-fed GEMM: Y[M,N] = X[M,K] @ W[N,K]^T + bias[N]
// Block: 256 thr = 8 waves (2M x 4N), wave tile 32x32, block tile 64x128.
// K consumed in KC=32 chunks, double-buffered in LDS via tensor_load_to_lds.
// grid = (M/64, N/128); K must be a multiple of KC.
// ---------------------------------------------------------------------------
__global__ void __launch_bounds__(256) gemm_xwT_bias_tdm(const float* __restrict__ X,
                                                         const float* __restrict__ W,
                                                         const float* __restrict__ bias,
                                                         float* __restrict__ Y,
                                                         int K, int N) {
  __shared__ float Asm[2][64 * LDS_STRIDE];    // 2 x 8448 B
  __shared__ float Bsm[2][128 * LDS_STRIDE];   // 2 x 16896 B

  const int lane = threadIdx.x & 31;
  const int wave = threadIdx.x >> 5;
  const int m0w  = (wave >> 2) * 32;   // wave tile origin inside block tile
  const int n0w  = (wave & 3) * 32;
  const int mblk = blockIdx.x * 64;
  const int nblk = blockIdx.y * 128;
  const int NC   = K / KC;

  unsigned int ldsA0 = (unsigned int)(uintptr_t)&Asm[0][0];
  unsigned int ldsA1 = (unsigned int)(uintptr_t)&Asm[1][0];
  unsigned int ldsB0 = (unsigned int)(uintptr_t)&Bsm[0][0];
  unsigned int ldsB1 = (unsigned int)(uintptr_t)&Bsm[1][0];

  v8f c00 = {}, c01 = {}, c10 = {}, c11 = {};

  if (wave == 0) {  // prologue: chunk 0 in flight
    tdm_load_2d(ldsA0, X + (size_t)mblk * K, K, 64, KC, 64, K);
    tdm_load_2d(ldsB0, W + (size_t)nblk * K, K, 128, KC, 128, K);
  }

  for (int c = 0; c < NC; ++c) {
    if (wave == 0) {
      if (c + 1 < NC) {
        const int kn = (c + 1) * KC;
        tdm_load_2d((c & 1) ? ldsA0 : ldsA1, X + (size_t)mblk * K + kn, K, 64, KC, 64, K);
        tdm_load_2d((c & 1) ? ldsB0 : ldsB1, W + (size_t)nblk * K + kn, K, 128, KC, 128, K);
        __builtin_amdgcn_s_wait_tensorcnt(2);   // chunk c landed; chunk c+1 in flight
      } else {
        __builtin_amdgcn_s_wait_tensorcnt(0);
      }
    }
    __syncthreads();   // publish LDS tile of chunk c to all waves

    const float* As = (c & 1) ? &Asm[1][0] : &Asm[0][0];
    const float* Bs = (c & 1) ? &Bsm[1][0] : &Bsm[0][0];
#pragma unroll
    for (int ks = 0; ks < KC; ks += 4) {
      v2f a0 = lds_frag(As, m0w,      ks, lane);
      v2f a1 = lds_frag(As, m0w + 16, ks, lane);
      v2f b0 = lds_frag(Bs, n0w,      ks, lane);
      v2f b1 = lds_frag(Bs, n0w + 16, ks, lane);
      c00 = WMMA_F32(a0, b0, c00);
      c01 = WMMA_F32(a0, b1, c01);
      c10 = WMMA_F32(a1, b0, c10);
      c11 = WMMA_F32(a1, b1, c11);
    }
    __syncthreads();   // all reads of buffer (c&1) done before TDM overwrites it
  }

  const int kh = lane >> 4;
  const int nc = lane & 15;
  const int m0 = mblk + m0w;
  const int n0 = nblk + n0w;
  const float bv0 = bias[n0 + nc];
  const float bv1 = bias[n0 + 16 + nc];
#pragma unroll
  for (int r = 0; r < 8; ++r) {
    int m = m0 + r + 8 * kh;
    Y[(size_t)m * N + n0 + nc]             = c00[r] + bv0;
    Y[(size_t)m * N + n0 + 16 + nc]        = c01[r] + bv1;
    Y[(size_t)(m + 16) * N + n0 + nc]      = c10[r] + bv0;
    Y[(size_t)(m + 16) * N + n0 + 16 + nc] = c11[r] + bv1;
  }
}

// ---------------------------------------------------------------------------
// Device-wide barrier for the persistent scan kernel (all blocks resident).
// ---------------------------------------------------------------------------
static __device__ __forceinline__ void gsync(int* cnt, int* gen, int nblocks) {
  __syncthreads();
  __threadfence();
  if (threadIdx.x == 0) {
    volatile int* vgen = (volatile int*)gen;
    int g = *vgen;
    if (atomicAdd(cnt, 1) == nblocks - 1) {
      *cnt = 0;
      __threadfence();
      atomicAdd(gen, 1);
    } else {
      while (*vgen == g) { __builtin_amdgcn_s_sleep(2); }
    }
  }
  __syncthreads();
  __threadfence();
}

__global__ void sync_init_kernel(int* cnt, int* gen) {
  if (threadIdx.x == 0) { *cnt = 0; *gen = 0; }
}

// ---------------------------------------------------------------------------
// Persistent GRU scan for one layer. Grid: 6 blocks x 256 thr = 48 waves.
// Per step: 3 GEMMs [32x512] = hcur @ W{r,z,h}^T as 48 wave-jobs of 32x32,
// then device barrier; elementwise gates; barrier. hcur/W are L2-hot.
// ---------------------------------------------------------------------------
__global__ void __launch_bounds__(256) gru_scan_kernel(
    const float* __restrict__ xr, const float* __restrict__ xz, const float* __restrict__ xh,
    const float* __restrict__ Whr, const float* __restrict__ Whz, const float* __restrict__ Whh,
    const float* __restrict__ h0,
    float* __restrict__ hcur, float* __restrict__ hr, float* __restrict__ hz,
    float* __restrict__ hh,
    float* __restrict__ hs, float* __restrict__ hfinal,
    int* sync_cnt, int* sync_gen, int S) {
  const int nb = gridDim.x;
  const int tid = blockIdx.x * blockDim.x + threadIdx.x;
  const int nthreads = gridDim.x * blockDim.x;
  const int lane = threadIdx.x & 31;

  for (int e = tid; e < 32 * 512; e += nthreads) hcur[e] = h0[e];
  gsync(sync_cnt, sync_gen, nb);

  const int gw  = blockIdx.x * (blockDim.x >> 5) + (threadIdx.x >> 5);  // 0..47
  const int mat = gw >> 4;
  const int n0  = (gw & 15) * 32;
  const float* __restrict__ W = (mat == 0) ? Whr : (mat == 1) ? Whz : Whh;
  float* __restrict__ O       = (mat == 0) ? hr  : (mat == 1) ? hz  : hh;

  for (int t = 0; t < S; ++t) {
    v8f c00 = {}, c01 = {}, c10 = {}, c11 = {};
    for (int k0 = 0; k0 < 512; k0 += 4) {
      v2f a0 = load_a_frag(hcur, 512, 0, k0, lane);
      v2f a1 = load_a_frag(hcur, 512, 16, k0, lane);
      v2f b0 = load_b_frag(W, 512, n0, k0, lane);
      v2f b1 = load_b_frag(W, 512, n0 + 16, k0, lane);
      c00 = WMMA_F32(a0, b0, c00);
      c01 = WMMA_F32(a0, b1, c01);
      c10 = WMMA_F32(a1, b0, c10);
      c11 = WMMA_F32(a1, b1, c11);
    }
    {
      const int kh = lane >> 4;
      const int nc = lane & 15;
#pragma unroll
      for (int r = 0; r < 8; ++r) {
        int m = r + 8 * kh;
        O[m * 512 + n0 + nc]             = c00[r];
        O[m * 512 + n0 + 16 + nc]        = c01[r];
        O[(m + 16) * 512 + n0 + nc]      = c10[r];
        O[(m + 16) * 512 + n0 + 16 + nc] = c11[r];
      }
    }
    gsync(sync_cnt, sync_gen, nb);

    const size_t toff = (size_t)t * 32 * 512;
    for (int e = tid; e < 32 * 512; e += nthreads) {
      float rg = sigmoidf_(xr[toff + e] + hr[e]);
      float zg = sigmoidf_(xz[toff + e] + hz[e]);
      float h1 = tanhf(xh[toff + e] + rg * hh[e]);
      float hn = (1.0f - zg) * h1 + zg * hcur[e];
      hcur[e] = hn;
      hs[toff + e] = hn;
      if (t == S - 1) hfinal[e] = hn;
    }
    gsync(sync_cnt, sync_gen, nb);
  }
}

// ---------------------------------------------------------------------------
extern "C" void kernel_launch(void* const* d_in, const int* in_sizes, int n_in,
                              void* d_out, int out_size, void* d_ws, size_t ws_size,
                              hipStream_t stream) {
  (void)in_sizes; (void)n_in; (void)out_size; (void)ws_size;

  const int*   inputs = (const int*)d_in[0];
  const float* hidden = (const float*)d_in[1];
  const float* emb    = (const float*)d_in[2];
  const float* Wir    = (const float*)d_in[3];
  const float* bir    = (const float*)d_in[4];
  const float* Wiz    = (const float*)d_in[5];
  const float* biz    = (const float*)d_in[6];
  const float* Wih    = (const float*)d_in[7];
  const float* bih    = (const float*)d_in[8];
  const float* Whr    = (const float*)d_in[9];
  const float* Whz    = (const float*)d_in[10];
  const float* Whh    = (const float*)d_in[11];
  const float* Wout   = (const float*)d_in[12];
  const float* bout   = (const float*)d_in[13];
  float* out = (float*)d_out;

  constexpr int S = 64, B = 32, H = 512, E = 512, V = 32000, L = 2;
  constexpr int M = S * B;  // 2048

  float* ws   = (float*)d_ws;
  float* bufA = ws;                              // [M,H] ping
  float* bufB = bufA + (size_t)M * H;            // [M,H] pong
  float* xr   = bufB + (size_t)M * H;            // [M,H]
  float* xz   = xr + (size_t)M * H;
  float* xh   = xz + (size_t)M * H;
  float* hr   = xh + (size_t)M * H;              // [B,H]
  float* hz   = hr + B * H;
  float* hh   = hz + B * H;
  float* hcur = hh + B * H;
  int*   sync = (int*)(hcur + B * H);

  float* logits = out;                           // [S,B,V]
  float* hfin   = out + (size_t)M * V;           // [L,B,H]

  embed_kernel<<<M, 256, 0, stream>>>(inputs, emb, bufA);

  for (int l = 0; l < L; ++l) {
    const float* xin = (l & 1) ? bufB : bufA;
    float* hsout     = (l & 1) ? bufA : bufB;
    dim3 g(M / 64, H / 128);
    gemm_xwT_bias_tdm<<<g, 256, 0, stream>>>(xin, Wir + (size_t)l * H * E, bir + (size_t)l * H, xr, E, H);
    gemm_xwT_bias_tdm<<<g, 256, 0, stream>>>(xin, Wiz + (size_t)l * H * E, biz + (size_t)l * H, xz, E, H);
    gemm_xwT_bias_tdm<<<g, 256, 0, stream>>>(xin, Wih + (size_t)l * H * E, bih + (size_t)l * H, xh, E, H);
    sync_init_kernel<<<1, 32, 0, stream>>>(sync, sync + 1);
    gru_scan_kernel<<<6, 256, 0, stream>>>(
        xr, xz, xh,
        Whr + (size_t)l * H * H, Whz + (size_t)l * H * H, Whh + (size_t)l * H * H,
        hidden + (size_t)l * B * H,
        hcur, hr, hz, hh, hsout, hfin + (size_t)l * B * H,
        sync, sync + 1, S);
  }

  const float* outs = (L & 1) ? bufB : bufA;  // L=2 -> bufA
  dim3 go(M / 64, V / 128);
  gemm_xwT_bias_tdm<<<go, 256, 0, stream>>>(outs, Wout, bout, logits, H, V);
}